// CapsuleLayer_82695300317885
// MI455X (gfx1250) — compile-verified
//
#include <hip/hip_runtime.h>
#include <hip/hip_bf16.h>

// CapsNet dynamic routing on gfx1250.
// 3 fused passes recompute u_hat via V_WMMA_F32_16X16X4_F32 in transposed
// form (rows = (j,d), cols = batch); LDS tiles staged by the Tensor Data
// Mover (TENSOR_LOAD_TO_LDS) with TDM padding reproducing the bank-conflict
// free layouts; s accumulated with global_atomic_add_f32.

typedef __attribute__((ext_vector_type(2))) float    v2f;
typedef __attribute__((ext_vector_type(8))) float    v8f;
typedef __attribute__((ext_vector_type(4))) unsigned u32x4;
typedef __attribute__((ext_vector_type(4))) int      i32x4;
typedef __attribute__((ext_vector_type(8))) int      i32x8;

#define BATCH   64
#define NIN     32768
#define DIN     8
#define NOUT    2
#define DOUT    16
#define ITILE   16
#define XROW    132               // padded dwords per batch row (128 data) -> conflict-free B loads
#define WROW    12                // padded dwords per (j,d) row (8 data)   -> conflict-free A loads
#define WTILE   (NOUT * DOUT * WROW)   // 384 dwords per i
#define SN      (BATCH * NOUT * DOUT)  // 2048 elements of s / v

#if defined(__HIP_DEVICE_COMPILE__) && __has_builtin(__builtin_amdgcn_tensor_load_to_lds) && __has_builtin(__builtin_amdgcn_s_wait_tensorcnt)
#define HAVE_TDM 1
#else
#define HAVE_TDM 0
#endif

__device__ __forceinline__ v8f wmma4(v2f a, v2f b, v8f c) {
  // D = A(16x4 f32) * B(4x16 f32) + C(16x16 f32)
  return __builtin_amdgcn_wmma_f32_16x16x4_f32(
      /*neg_a=*/false, a, /*neg_b=*/false, b,
      /*c_mod=*/(short)0, c, /*reuse_a=*/false, /*reuse_b=*/false);
}

#if HAVE_TDM
// 2D TDM tile load: tile_d0 contiguous elements per row, tile_d1 rows with
// row stride t_d0_stride (elements); LDS destination padded by pad_amount+1
// dwords every 2^(pad_interval+1) dwords. Descriptor per CDNA5 ISA 8.3-8.5.
__device__ __forceinline__ void tdm_load_2d(
    const void* gptr, unsigned lds_off, unsigned tile_d0, unsigned tile_d1,
    unsigned long long t_d0_stride, unsigned pad_interval, unsigned pad_amount) {
  unsigned long long ga = (unsigned long long)(uintptr_t)gptr;
  u32x4 g0;
  g0[0] = 1u;                                            // count=1 (valid), user mode
  g0[1] = lds_off;                                       // lds_addr (bytes)
  g0[2] = (unsigned)(ga & 0xFFFFFFFFu);                  // global_addr[31:0]
  g0[3] = (unsigned)((ga >> 32) & 0x01FFFFFFu) | (2u << 30);  // global_addr[56:32] | type=2
  i32x8 g1;
  g1[0] = (int)((2u << 16)                               // data_size = 4 bytes
              | (1u << 20)                               // pad_enable
              | (pad_interval << 22) | (pad_amount << 25));
  g1[1] = (int)((tile_d0 & 0xFFFFu) << 16);              // tensor_dim0[15:0] (== tile_d0)
  g1[2] = (int)(((tile_d0 >> 16) & 0xFFFFu) | ((tile_d1 & 0xFFFFu) << 16));  // dim0 hi | dim1 lo
  g1[3] = (int)(((tile_d1 >> 16) & 0xFFFFu) | ((tile_d0 & 0xFFFFu) << 16));  // dim1 hi | tile_dim0
  g1[4] = (int)(tile_d1 & 0xFFFFu);                      // tile_dim1, tile_dim2=0
  g1[5] = (int)(unsigned)(t_d0_stride & 0xFFFFFFFFull);  // tensor_dim0_stride lo
  g1[6] = (int)(unsigned)((t_d0_stride >> 32) & 0xFFFFull);  // stride hi, dim1_stride=0
  g1[7] = 0;
  i32x4 z4 = {0, 0, 0, 0};
#if __clang_major__ >= 23
  i32x8 z8 = {0, 0, 0, 0, 0, 0, 0, 0};
  __builtin_amdgcn_tensor_load_to_lds(g0, g1, z4, z4, z8, 0);
#else
  __builtin_amdgcn_tensor_load_to_lds(g0, g1, z4, z4, 0);
#endif
}
#endif

// One routing pass over this block's i-tile:
//   s[b,j,d] += softmax_j(u_hat . vin)[b,i,j] * u_hat[b,i,j,d]
__global__ __launch_bounds__(128) void caps_pass(
    const float* __restrict__ x, const float* __restrict__ Wm,
    const float* __restrict__ vin, float* __restrict__ sout) {
  __shared__ float lx[BATCH * XROW];   // x tile: [b][ii*8+k], padded rows of 132 dw
  __shared__ float lw[ITILE * WTILE];  // W tile: [ii][(j*16+d)*12 + k], padded rows of 12 dw

  const int tid  = threadIdx.x;
  const int i0   = blockIdx.x * ITILE;
  const int lane = tid & 31;
  const int wid  = tid >> 5;             // batch tile: [wid*16, wid*16+16)
  const int nlo  = lane & 15;            // A: M row (j,d) / B: N col (batch)
  const int klo  = (lane & 16) ? 2 : 0;  // K lane-half split per ISA f32 layouts
  const int dhi  = (lane & 16) ? 8 : 0;  // C-layout: high lanes hold M rows +8 (d offset)
  const int bb   = wid * 16 + nlo;       // this lane's batch

#if HAVE_TDM
  if (wid == 0) {
    // x tile: 64 rows x 128 dw, row stride NIN*8 elems; pad 4 dw / 128 dw -> XROW=132
    tdm_load_2d(x + (size_t)i0 * DIN, (unsigned)(uintptr_t)lx,
                ITILE * DIN, BATCH, (unsigned long long)NIN * DIN, 6u, 3u);
    // W tile: 16 rows x 256 dw contiguous; pad 4 dw / 8 dw -> WROW=12, WTILE=384
    tdm_load_2d(Wm + (size_t)i0 * (NOUT * DOUT * DIN), (unsigned)(uintptr_t)lw,
                NOUT * DOUT * DIN, ITILE, (unsigned long long)(NOUT * DOUT * DIN), 2u, 3u);
    __builtin_amdgcn_s_wait_tensorcnt(0);
  }
#else
  for (int idx = tid; idx < BATCH * 32; idx += 128) {
    int b = idx >> 5, q = idx & 31;
    const float4 v = *(const float4*)(x + ((size_t)b * NIN + i0) * DIN + q * 4);
    *(float4*)(lx + b * XROW + q * 4) = v;
  }
  for (int idx = tid; idx < ITILE * 128; idx += 128) {
    int ii = idx >> 7, r2 = idx & 127;
    int jd = r2 >> 2, k2 = (r2 & 3) * 2;
    const v2f v = *(const v2f*)(Wm + (size_t)(i0 + ii) * (NOUT * DOUT * DIN) + jd * DIN + k2);
    *(v2f*)(lw + ii * WTILE + jd * WROW + k2) = v;
  }
#endif

  // ---- preload routing v in transposed-C register layout: vT[j][r] = v[bb, j, r+dhi] ----
  float vT[2][8];
#pragma unroll
  for (int r = 0; r < 8; ++r) {
    vT[0][r] = vin[(bb * NOUT + 0) * DOUT + r + dhi];
    vT[1][r] = vin[(bb * NOUT + 1) * DOUT + r + dhi];
  }
  __syncthreads();

  v8f acc0 = {0.f, 0.f, 0.f, 0.f, 0.f, 0.f, 0.f, 0.f};  // s[bb, 0, r+dhi]
  v8f acc1 = acc0;                                       // s[bb, 1, r+dhi]
  const float* lwp0 = lw + (0 * DOUT + nlo) * WROW + klo;  // A rows: j=0, d=nlo
  const float* lwp1 = lw + (1 * DOUT + nlo) * WROW + klo;  // A rows: j=1, d=nlo
  const float* lxp  = lx + bb * XROW + klo;                // B cols: batch bb

  for (int ii = 0; ii < ITILE; ++ii) {
    const v2f aw00 = *(const v2f*)(lwp0 + ii * WTILE);       // W k=klo..klo+1
    const v2f aw10 = *(const v2f*)(lwp0 + ii * WTILE + 4);   // W k=4+klo..
    const v2f aw01 = *(const v2f*)(lwp1 + ii * WTILE);
    const v2f aw11 = *(const v2f*)(lwp1 + ii * WTILE + 4);
    const v2f bx0  = *(const v2f*)(lxp + ii * DIN);          // x k=klo..klo+1
    const v2f bx1  = *(const v2f*)(lxp + ii * DIN + 4);      // x k=4+klo..

    v8f cz = {0.f, 0.f, 0.f, 0.f, 0.f, 0.f, 0.f, 0.f};
    v8f c0 = wmma4(aw00, bx0, cz); c0 = wmma4(aw10, bx1, c0);  // u_hatT[(0,d), b]
    v8f c1 = wmma4(aw01, bx0, cz); c1 = wmma4(aw11, bx1, c1);  // u_hatT[(1,d), b]

    // logits: in-lane reduce over 8 d-rows, then one xor-16 to join d halves
    float p0 = c0[0] * vT[0][0];
    float p1 = c1[0] * vT[1][0];
#pragma unroll
    for (int r = 1; r < 8; ++r) {
      p0 = fmaf(c0[r], vT[0][r], p0);
      p1 = fmaf(c1[r], vT[1][r], p1);
    }
    p0 += __shfl_xor(p0, 16, 32);   // full logit(bb, j=0) in both lane copies
    p1 += __shfl_xor(p1, 16, 32);   // full logit(bb, j=1)

    float mx  = fmaxf(p0, p1);
    float e0  = __expf(p0 - mx), e1 = __expf(p1 - mx);
    float inv = 1.0f / (e0 + e1);
    float cw0 = e0 * inv, cw1 = e1 * inv;   // softmax coupling coefficients

#pragma unroll
    for (int r = 0; r < 8; ++r) {           // s += c * u_hat
      acc0[r] += cw0 * c0[r];
      acc1[r] += cw1 * c1[r];
    }
  }

  // ---- flush per-wave s partials (distinct (b,j,d) per lane slot) ----
#pragma unroll
  for (int r = 0; r < 8; ++r) {
    unsafeAtomicAdd(&sout[(bb * NOUT + 0) * DOUT + r + dhi], acc0[r]);
    unsafeAtomicAdd(&sout[(bb * NOUT + 1) * DOUT + r + dhi], acc1[r]);
  }
}

// v = squash(s); mode 0: vacc = v; mode 1: vacc += v; mode 2: out = v.
// Also re-zeroes s for the next pass.
__global__ __launch_bounds__(128) void caps_squash(
    float* __restrict__ s, float* __restrict__ vacc, float* __restrict__ out, int mode) {
  int t = threadIdx.x;  // one (b,j) pair per thread, 128 total
  float sv[DOUT];
  float s2 = 0.f;
  int base = t * DOUT;
#pragma unroll
  for (int d = 0; d < DOUT; ++d) { sv[d] = s[base + d]; s2 += sv[d] * sv[d]; }
  float scale = (s2 / (1.0f + s2)) * __frsqrt_rn(s2 + 1e-7f);
#pragma unroll
  for (int d = 0; d < DOUT; ++d) {
    float v = sv[d] * scale;
    s[base + d] = 0.f;
    if (mode == 0)      vacc[base + d] = v;
    else if (mode == 1) vacc[base + d] += v;
    else                out[base + d] = v;
  }
}

__global__ __launch_bounds__(256) void caps_init(float* __restrict__ p, int n) {
  for (int i = threadIdx.x; i < n; i += 256) p[i] = 0.f;
}

extern "C" void kernel_launch(void* const* d_in, const int* in_sizes, int n_in,
                              void* d_out, int out_size, void* d_ws, size_t ws_size,
                              hipStream_t stream) {
  const float* x  = (const float*)d_in[0];  // [64, 32768, 8]
  const float* Wm = (const float*)d_in[1];  // [1, 32768, 2, 16, 8]
  float* out  = (float*)d_out;              // [64, 2, 16]
  float* s    = (float*)d_ws;               // 2048 floats
  float* vacc = s + SN;                     // 2048 floats (routing v, accumulated)

  const int grid = NIN / ITILE;  // 2048 blocks

  caps_init<<<1, 256, 0, stream>>>(s, 2 * SN);                 // s = 0, vacc = 0 (iter0: softmax(0)=0.5)
  caps_pass<<<grid, 128, 0, stream>>>(x, Wm, vacc, s);         // iter 0
  caps_squash<<<1, 128, 0, stream>>>(s, vacc, out, 0);         // vacc = v1, s = 0
  caps_pass<<<grid, 128, 0, stream>>>(x, Wm, vacc, s);         // iter 1 (logits = u.v1)
  caps_squash<<<1, 128, 0, stream>>>(s, vacc, out, 1);         // vacc = v1+v2, s = 0
  caps_pass<<<grid, 128, 0, stream>>>(x, Wm, vacc, s);         // iter 2 (logits = u.(v1+v2))
  caps_squash<<<1, 128, 0, stream>>>(s, vacc, out, 2);         // out = v3
}